// GatherNd_8890582303354
// MI455X (gfx1250) — compile-verified
//
#include <hip/hip_runtime.h>

// GatherNd, batch_dims=1:  out[b, j, :] = data[b, indices[b, j, 0], :]
// data:    [B=8, N=50000, D=256] float32
// indices: [B=8, M=65536, 1]     int64
// out:     [B=8, M=65536, D=256] float32
//
// Pure memory-bound gather: ~1 GB moved, 0 FLOPs -> HBM roofline ~45us.
// One wave32 per 1KB output row; b128 loads/stores; wave-uniform scalar
// index load (s_load_b64); non-temporal output stores so the streamed
// 512MB output doesn't evict the gather working set (51MB/batch) from L2.
// M is a power of two -> batch id via shift, no 64-bit division.

typedef float v4f __attribute__((ext_vector_type(4)));

constexpr int kD            = 256;   // row length in floats (1 KB)
constexpr int kDShift       = 8;     // log2(kD)
constexpr int kRowsPerBlock = 8;     // 8 waves of 32 lanes per block

__global__ __launch_bounds__(256) void gather_rows_kernel(
    const float* __restrict__ data,
    const long long* __restrict__ indices,
    float* __restrict__ out,
    int N, int mShift, long long rows)
{
    const int lane = threadIdx.x & 31;
    // Force wave id into an SGPR: everything derived from it (notably the
    // index address) is wave-uniform, so the i64 index load becomes SMEM.
    const int wave = __builtin_amdgcn_readfirstlane((int)(threadIdx.x >> 5));

    const long long row = (long long)blockIdx.x * kRowsPerBlock + wave;
    if (row >= rows) return;

    const int b = (int)(row >> mShift);      // batch id: M = 2^mShift
    const long long idx = indices[row];      // wave-uniform -> s_load_b64

    const v4f* __restrict__ src =
        (const v4f*)(data + (((long long)b * N + idx) << kDShift));
    v4f* __restrict__ dst = (v4f*)(out + (row << kDShift));

    // 256 floats = 64 x float4; 32 lanes x 2 vectors. Both b128 loads
    // issue before the stores so they pipeline.
    v4f v0 = src[lane];
    v4f v1 = src[lane + 32];

    // Output is write-once/never-reread: TH=NT at both cache levels keeps
    // L2 capacity for the gathered data rows (which are re-read ~1.3x).
    __builtin_nontemporal_store(v0, dst + lane);
    __builtin_nontemporal_store(v1, dst + lane + 32);
}

extern "C" void kernel_launch(void* const* d_in, const int* in_sizes, int n_in,
                              void* d_out, int out_size, void* d_ws, size_t ws_size,
                              hipStream_t stream) {
    const float*      data    = (const float*)d_in[0];
    const long long*  indices = (const long long*)d_in[1];
    float*            out     = (float*)d_out;

    const int B = 8;
    const long long rows = (long long)in_sizes[1];                     // B*M
    const long long M    = rows / B;                                   // 65536
    const int N = (int)((long long)in_sizes[0] / ((long long)B * kD)); // 50000

    // M is a power of two for this problem; compute its log2 on the host.
    int mShift = 0;
    while ((1LL << mShift) < M) ++mShift;

    const unsigned blocks = (unsigned)((rows + kRowsPerBlock - 1) / kRowsPerBlock);
    gather_rows_kernel<<<dim3(blocks), dim3(256), 0, stream>>>(
        data, indices, out, N, mShift, rows);
}